// CRFTokenClassifier_16870631539478
// MI455X (gfx1250) — compile-verified
//
#include <hip/hip_runtime.h>

// CRF token classifier loss for MI455X (gfx1250, wave32, WMMA).
// Stage 1: emissions[B*S, 3] = hidden[B*S, 768] @ W[768, 3] + b  via V_WMMA_F32_16X16X4_F32
//          B operand is staged zero-padded in LDS so the inner loop is branch-free:
//          one global_load_b64 (A) + one ds_load_b64 (B) per WMMA.
// Stage 2: CRF gold score + forward algorithm (L=3) per batch lane, reduce to -mean.

#define B_ 64
#define S_ 512
#define H_ 768
#define L_ 3
#define M_ (B_ * S_)          // 32768 rows
#define WAVES_PER_BLOCK 8
#define TILES (M_ / 16)       // 2048

typedef __attribute__((ext_vector_type(2))) float v2f;
typedef __attribute__((ext_vector_type(8))) float v8f;

__global__ __launch_bounds__(256) void emissions_wmma_kernel(
    const float* __restrict__ hidden,   // [M_, H_]
    const float* __restrict__ W,        // [H_, L_]
    const float* __restrict__ bias,     // [L_]
    float* __restrict__ em)             // [M_, L_]
{
    // Padded B-operand staging: blds[p][n] = { W[2p][n], W[2p+1][n] }, zero for n >= 3.
    // One ds_load_b64 per lane per WMMA, no EXEC masking in the hot loop.
    __shared__ v2f blds[(H_ / 2) * 16];                 // 48 KB of 320 KB WGP LDS
    const int tid = threadIdx.x;
    for (int idx = tid; idx < (H_ / 2) * 16; idx += 256) {
        const int p = idx >> 4;
        const int n = idx & 15;
        v2f v = {0.0f, 0.0f};
        if (n < L_) {
            v[0] = W[(2 * p)     * L_ + n];
            v[1] = W[(2 * p + 1) * L_ + n];
        }
        blds[idx] = v;
    }
    __syncthreads();

    const int wave = tid >> 5;
    const int lane = tid & 31;
    const int n    = lane & 15;         // column index (matrix N / A-row M)
    const int half = lane >> 4;         // 0 or 1
    const int kk   = half * 2;          // K sub-offset held by this half-wave

    const int tile = blockIdx.x * WAVES_PER_BLOCK + wave;   // 0..2047
    const float* hrow = hidden + (tile * 16 + n) * H_;      // 32-bit indexing
    const int bidx_base = half * 16 + n;

    v8f c = {};
    #pragma unroll 8
    for (int k = 0; k < H_; k += 4) {
        // A (16x4 f32): lane holds A[n][k+kk], A[n][k+kk+1] -> contiguous 8B load
        v2f a = *(const v2f*)(hrow + k + kk);
        // B (4x16 f32): lane holds W[k+kk][n], W[k+kk+1][n] (pre-padded in LDS)
        v2f bb = blds[(k >> 1) * 16 + bidx_base];
        c = __builtin_amdgcn_wmma_f32_16x16x4_f32(
                /*neg_a=*/false, a, /*neg_b=*/false, bb,
                /*c_mod=*/(short)0, c, /*reuse_a=*/false, /*reuse_b=*/false);
    }

    // C/D layout: VGPR v -> row (v + 8*half) of the tile, col = n
    if (n < L_) {
        const float bn = bias[n];
        #pragma unroll
        for (int v = 0; v < 8; ++v) {
            const int row = tile * 16 + v + 8 * half;
            em[row * L_ + n] = c[v] + bn;
        }
    }
}

__global__ __launch_bounds__(B_) void crf_nll_kernel(
    const float* __restrict__ em,       // [B_, S_, L_]
    const int*   __restrict__ labels,   // [B_, S_]
    const int*   __restrict__ amask,    // [B_, S_]
    const float* __restrict__ startT,   // [L_]
    const float* __restrict__ endT,     // [L_]
    const float* __restrict__ trans,    // [L_, L_]
    float* __restrict__ out)            // [1]
{
    __shared__ float red[B_];
    const int b = threadIdx.x;

    float st[L_], en[L_], tr[L_][L_];
    #pragma unroll
    for (int i = 0; i < L_; ++i) {
        st[i] = startT[i];
        en[i] = endT[i];
        #pragma unroll
        for (int j = 0; j < L_; ++j) tr[i][j] = trans[i * L_ + j];
    }

    const float* e  = em + b * S_ * L_;
    const int* lab  = labels + b * S_;
    const int* mk   = amask  + b * S_;

    int tag0 = lab[0]; if (tag0 < 0) tag0 = 0;
    float score = st[tag0] + e[tag0];
    int prev = tag0, last = tag0;

    float alpha[L_];
    #pragma unroll
    for (int l = 0; l < L_; ++l) alpha[l] = st[l] + e[l];

    for (int s = 1; s < S_; ++s) {
        const float* es = e + s * L_;
        int tg = lab[s]; if (tg < 0) tg = 0;
        float m = (float)mk[s];

        // gold-path score
        score += (es[tg] + tr[prev][tg]) * m;
        last = (m > 0.0f) ? tg : last;
        prev = tg;

        // forward algorithm: alpha_next[j] = LSE_i(alpha[i] + tr[i][j]) + es[j]
        float nxt[L_];
        #pragma unroll
        for (int j = 0; j < L_; ++j) {
            float x0 = alpha[0] + tr[0][j];
            float x1 = alpha[1] + tr[1][j];
            float x2 = alpha[2] + tr[2][j];
            float mx = fmaxf(x0, fmaxf(x1, x2));
            nxt[j] = mx + logf(expf(x0 - mx) + expf(x1 - mx) + expf(x2 - mx))
                        + es[j];
        }
        const bool upd = m > 0.0f;
        #pragma unroll
        for (int j = 0; j < L_; ++j) alpha[j] = upd ? nxt[j] : alpha[j];
    }

    score += en[last];
    {
        float x0 = alpha[0] + en[0];
        float x1 = alpha[1] + en[1];
        float x2 = alpha[2] + en[2];
        float mx = fmaxf(x0, fmaxf(x1, x2));
        float logz = mx + logf(expf(x0 - mx) + expf(x1 - mx) + expf(x2 - mx));
        red[b] = score - logz;
    }
    __syncthreads();
    if (b == 0) {
        float s = 0.0f;
        #pragma unroll
        for (int i = 0; i < B_; ++i) s += red[i];
        out[0] = -s / (float)B_;
    }
}

extern "C" void kernel_launch(void* const* d_in, const int* in_sizes, int n_in,
                              void* d_out, int out_size, void* d_ws, size_t ws_size,
                              hipStream_t stream) {
    const float* hidden = (const float*)d_in[0];
    const float* W      = (const float*)d_in[1];
    const float* bias   = (const float*)d_in[2];
    const float* startT = (const float*)d_in[3];
    const float* endT   = (const float*)d_in[4];
    const float* trans  = (const float*)d_in[5];
    const int*   amask  = (const int*)d_in[6];
    const int*   labels = (const int*)d_in[7];
    float* out = (float*)d_out;
    float* em  = (float*)d_ws;          // M_ * L_ * 4 = 384 KiB scratch

    (void)in_sizes; (void)n_in; (void)out_size; (void)ws_size;

    emissions_wmma_kernel<<<dim3(TILES / WAVES_PER_BLOCK), dim3(256), 0, stream>>>(
        hidden, W, bias, em);
    crf_nll_kernel<<<dim3(1), dim3(B_), 0, stream>>>(
        em, labels, amask, startT, endT, trans, out);
}